// GAT_35708358099579
// MI455X (gfx1250) — compile-verified
//
#include <hip/hip_runtime.h>
#include <math.h>

#define F_DIM 128
#define NHEAD 4
#define DHEAD 32

typedef __attribute__((ext_vector_type(2))) float v2f;
typedef __attribute__((ext_vector_type(8))) float v8f;
typedef __attribute__((ext_vector_type(4))) int   v4i;

typedef __attribute__((address_space(1))) v4i* gptr_v4i;
typedef __attribute__((address_space(3))) v4i* lptr_v4i;

#if defined(__has_builtin)
#if __has_builtin(__builtin_amdgcn_global_load_async_to_lds_b128)
#define HAVE_ASYNC_LDS 1
#endif
#if __has_builtin(__builtin_amdgcn_s_wait_asynccnt)
#define HAVE_WAIT_ASYNC_BUILTIN 1
#endif
#endif

__device__ inline void wait_asynccnt0() {
#if defined(HAVE_WAIT_ASYNC_BUILTIN)
  __builtin_amdgcn_s_wait_asynccnt(0);
#else
  asm volatile("s_wait_asynccnt 0x0" ::: "memory");
#endif
}

// ---------------------------------------------------------------------------
// feat = X (nrows x 128) @ W (128 x 128), f32 WMMA 16x16x4.
// One block = 8 waves; each wave owns one 16x16 output tile of the 16x128
// row-stripe. X tile staged into LDS via async DMA (stride 132 floats keeps
// 16B chunks aligned and A-fragment reads bank-conflict free).
// ---------------------------------------------------------------------------
__global__ __launch_bounds__(256) void gat_gemm_wmma(
    const float* __restrict__ X, const float* __restrict__ W,
    float* __restrict__ out, int nrows) {
  __shared__ float sx[16 * 132];
  const int tile_row = blockIdx.x * 16;

#if defined(HAVE_ASYNC_LDS)
  // 16 rows x 32 chunks of 16B each = 512 chunks, 2 per thread.
  #pragma unroll
  for (int c = threadIdx.x; c < 512; c += 256) {
    int r = c >> 5, q = (c & 31) << 2;          // row, float-column
    int gr = tile_row + r;
    if (gr >= nrows) gr = nrows - 1;            // clamp (pad rows unused)
    const float* gp = X + (size_t)gr * F_DIM + q;
    __builtin_amdgcn_global_load_async_to_lds_b128(
        (gptr_v4i)(float*)gp,
        (lptr_v4i)&sx[r * 132 + q],
        0, 0);
  }
  wait_asynccnt0();
  __syncthreads();
#else
  for (int i = threadIdx.x; i < 16 * F_DIM; i += 256) {
    int r = i >> 7, c = i & 127;
    int gr = tile_row + r;
    sx[r * 132 + c] = (gr < nrows) ? X[(size_t)gr * F_DIM + c] : 0.f;
  }
  __syncthreads();
#endif

  const int wave  = threadIdx.x >> 5;
  const int lane  = threadIdx.x & 31;
  const int m     = lane & 15;          // A row / B-C column-within-tile
  const int khalf = (lane >> 4) << 1;   // lanes 0-15 -> K+0/1, lanes 16-31 -> K+2/3
  const int ncol  = wave * 16 + m;      // global output column

  v8f acc = {};
  #pragma unroll 4
  for (int k0 = 0; k0 < F_DIM; k0 += 4) {
    v2f a, b;
    a.x = sx[m * 132 + k0 + khalf];
    a.y = sx[m * 132 + k0 + khalf + 1];
    b.x = W[(k0 + khalf) * F_DIM + ncol];
    b.y = W[(k0 + khalf + 1) * F_DIM + ncol];
    acc = __builtin_amdgcn_wmma_f32_16x16x4_f32(
        /*neg_a=*/false, a, /*neg_b=*/false, b,
        /*c_mod=*/(short)0, acc, /*reuse_a=*/false, /*reuse_b=*/false);
  }

  // C/D layout: VGPR j, lanes 0-15 -> M=j, lanes 16-31 -> M=8+j, N=lane&15
  const int row0 = tile_row + ((lane >> 4) << 3);
  float* op = out + (size_t)row0 * F_DIM + ncol;
  if (row0 + 8 <= nrows) {              // full tile: branch-free stores
    #pragma unroll
    for (int j = 0; j < 8; ++j) op[(size_t)j * F_DIM] = acc[j];
  } else {
    for (int j = 0; j < 8; ++j)
      if (row0 + j < nrows) op[(size_t)j * F_DIM] = acc[j];
  }
}

// ---------------------------------------------------------------------------
// el/er per (node, head); also initializes segment-max / segment-sum buffers.
// ---------------------------------------------------------------------------
__global__ void gat_attn_coef(const float* __restrict__ feat,
                              const float* __restrict__ al,
                              const float* __restrict__ ar,
                              float* __restrict__ el, float* __restrict__ er,
                              float* __restrict__ mbuf, float* __restrict__ sbuf,
                              int nh) {
  int idx = blockIdx.x * blockDim.x + threadIdx.x;
  if (idx >= nh) return;
  int n = idx >> 2, h = idx & 3;
  const float* f  = feat + (size_t)n * F_DIM + h * DHEAD;
  const float* a1 = al + h * DHEAD;
  const float* a2 = ar + h * DHEAD;
  float sl = 0.f, sr = 0.f;
  #pragma unroll
  for (int d = 0; d < DHEAD; ++d) {
    float fv = f[d];
    sl += fv * a1[d];
    sr += fv * a2[d];
  }
  el[idx] = sl;
  er[idx] = sr;
  mbuf[idx] = -__builtin_huge_valf();
  sbuf[idx] = 0.f;
}

// Sign-aware int-punned float atomic max (monotone under IEEE bit ordering).
__device__ inline void atomic_max_f32(float* addr, float val) {
  if (val >= 0.f)
    atomicMax((int*)addr, __float_as_int(val));
  else
    atomicMin((unsigned int*)addr, (unsigned int)__float_as_int(val));
}

// ---------------------------------------------------------------------------
// e = leaky_relu(el[src]+er[dst], 0.2); segment max into mbuf[dst].
// ---------------------------------------------------------------------------
__global__ void gat_edge_score(const int* __restrict__ src, const int* __restrict__ dst,
                               const float* __restrict__ el, const float* __restrict__ er,
                               float* __restrict__ score, float* __restrict__ mbuf,
                               int eh) {
  int idx = blockIdx.x * blockDim.x + threadIdx.x;
  if (idx >= eh) return;
  int e = idx >> 2, h = idx & 3;
  int u = src[e], v = dst[e];
  float s = el[u * NHEAD + h] + er[v * NHEAD + h];
  s = (s > 0.f) ? s : 0.2f * s;
  score[idx] = s;
  atomic_max_f32(&mbuf[v * NHEAD + h], s);
}

// ---------------------------------------------------------------------------
// ex = exp(e - m[dst]) in place; segment sum into sbuf[dst].
// ---------------------------------------------------------------------------
__global__ void gat_edge_exp(const int* __restrict__ dst,
                             float* __restrict__ score,
                             const float* __restrict__ mbuf,
                             float* __restrict__ sbuf, int eh) {
  int idx = blockIdx.x * blockDim.x + threadIdx.x;
  if (idx >= eh) return;
  int e = idx >> 2, h = idx & 3;
  int v = dst[e];
  float ex = __expf(score[idx] - mbuf[v * NHEAD + h]);
  score[idx] = ex;
  atomicAdd(&sbuf[v * NHEAD + h], ex);
}

// ---------------------------------------------------------------------------
// out[dst] += (ex/s[dst]) * feat[src]; one wave per edge, 4 floats per lane.
// ---------------------------------------------------------------------------
__global__ __launch_bounds__(256) void gat_edge_aggregate(
    const int* __restrict__ src, const int* __restrict__ dst,
    const float* __restrict__ ex, const float* __restrict__ sbuf,
    const float* __restrict__ feat, float* __restrict__ out, int ne) {
  int wave = threadIdx.x >> 5;
  int lane = threadIdx.x & 31;
  int e = blockIdx.x * 8 + wave;
  if (e >= ne) return;
  int u = src[e], v = dst[e];
  int h = lane >> 3;                        // lane*4 / 32
  float a = ex[e * NHEAD + h] / sbuf[v * NHEAD + h];
  const float4* fp = (const float4*)(feat + (size_t)u * F_DIM);
  float4 f = fp[lane];
  float* op = out + (size_t)v * F_DIM + lane * 4;
  atomicAdd(op + 0, a * f.x);
  atomicAdd(op + 1, a * f.y);
  atomicAdd(op + 2, a * f.z);
  atomicAdd(op + 3, a * f.w);
}

// ---------------------------------------------------------------------------
// in-place bias add + optional leaky_relu activation.
// ---------------------------------------------------------------------------
__global__ void gat_bias_act(float* __restrict__ buf, const float* __restrict__ bias,
                             float slope, int apply_act, int total) {
  int idx = blockIdx.x * blockDim.x + threadIdx.x;
  if (idx >= total) return;
  float v = buf[idx] + bias[idx & (F_DIM - 1)];
  if (apply_act) v = (v > 0.f) ? v : slope * v;
  buf[idx] = v;
}

__global__ void gat_zero(float* __restrict__ buf, int total) {
  int idx = blockIdx.x * blockDim.x + threadIdx.x;
  if (idx < total) buf[idx] = 0.f;
}

// ---------------------------------------------------------------------------
static void run_gat_layer(const float* xin, const int* src, const int* dst,
                          const float* Wm, const float* al, const float* ar,
                          const float* bias,
                          float* feat, float* el, float* er,
                          float* mbuf, float* sbuf, float* score,
                          float* outbuf, int N, int E,
                          float act_slope, int apply_act, hipStream_t stream) {
  const int total = N * F_DIM;
  gat_zero<<<(total + 255) / 256, 256, 0, stream>>>(outbuf, total);
  gat_gemm_wmma<<<(N + 15) / 16, 256, 0, stream>>>(xin, Wm, feat, N);
  gat_attn_coef<<<(N * NHEAD + 255) / 256, 256, 0, stream>>>(
      feat, al, ar, el, er, mbuf, sbuf, N * NHEAD);
  gat_edge_score<<<(E * NHEAD + 255) / 256, 256, 0, stream>>>(
      src, dst, el, er, score, mbuf, E * NHEAD);
  gat_edge_exp<<<(E * NHEAD + 255) / 256, 256, 0, stream>>>(
      dst, score, mbuf, sbuf, E * NHEAD);
  gat_edge_aggregate<<<(E + 7) / 8, 256, 0, stream>>>(
      src, dst, score, sbuf, feat, outbuf, E);
  gat_bias_act<<<(total + 255) / 256, 256, 0, stream>>>(
      outbuf, bias, act_slope, apply_act, total);
}

extern "C" void kernel_launch(void* const* d_in, const int* in_sizes, int n_in,
                              void* d_out, int out_size, void* d_ws, size_t ws_size,
                              hipStream_t stream) {
  const float* x   = (const float*)d_in[0];
  const int*   src = (const int*)d_in[1];
  const int*   dst = (const int*)d_in[2];
  const float* W1  = (const float*)d_in[3];
  const float* al1 = (const float*)d_in[4];
  const float* ar1 = (const float*)d_in[5];
  const float* b1  = (const float*)d_in[6];
  const float* W2  = (const float*)d_in[7];
  const float* al2 = (const float*)d_in[8];
  const float* ar2 = (const float*)d_in[9];
  const float* b2  = (const float*)d_in[10];

  const int N = in_sizes[0] / F_DIM;
  const int E = in_sizes[1];

  // Carve workspace (256B aligned slices).
  char* w = (char*)d_ws;
  size_t off = 0;
  auto take = [&](size_t elems) {
    float* p = (float*)(w + off);
    off = (off + elems * sizeof(float) + 255) & ~(size_t)255;
    return p;
  };
  float* feat  = take((size_t)N * F_DIM);   // per-layer x@W
  float* hbuf  = take((size_t)N * F_DIM);   // layer-1 output
  float* el    = take((size_t)N * NHEAD);
  float* er    = take((size_t)N * NHEAD);
  float* mbuf  = take((size_t)N * NHEAD);
  float* sbuf  = take((size_t)N * NHEAD);
  float* score = take((size_t)E * NHEAD);
  (void)ws_size; (void)n_in; (void)out_size;

  // Layer 1: activation leaky_relu(0.01)
  run_gat_layer(x, src, dst, W1, al1, ar1, b1,
                feat, el, er, mbuf, sbuf, score,
                hbuf, N, E, 0.01f, 1, stream);
  // Layer 2: no activation, straight into d_out
  run_gat_layer(hbuf, src, dst, W2, al2, ar2, b2,
                feat, el, er, mbuf, sbuf, score,
                (float*)d_out, N, E, 0.f, 0, stream);
}